// RoutingMaskLayer_30640296689906
// MI455X (gfx1250) — compile-verified
//
#include <hip/hip_runtime.h>

// RoutingMaskLayer: out[b,h,w,c] = in[b,h,w, argmax(routing[b,:])*128 + c]
// Shapes fixed by the reference: in [64,32,32,512] f32, routing [64,4] f32,
// out [64,32,32,128] f32.  Pure HBM-bandwidth op (~67 MB total traffic
// -> ~2.9 us at 23.3 TB/s).  Strategy: read only the selected 128-channel
// block, move it with the CDNA5 async Global<->LDS DMA path (b128, ASYNCcnt),
// fully coalesced, 2048 blocks to saturate the WGPs.

#define ROUTES      4
#define Bn          64
#define HW          1024            // 32*32
#define Cc          512
#define RW          128             // Cc / ROUTES
// float4 units
#define C4          (Cc / 4)        // 128 float4 per (b,h,w) input row
#define RW4         (RW / 4)        // 32 float4 per (b,h,w) output row
#define OUT4_PER_B  (HW * RW4)      // 32768
#define IN4_PER_B   (HW * C4)       // 131072

#define BLOCK   256                 // 8 waves (wave32)
#define CHUNKS  32                  // blocks per batch -> 2048 blocks total
#define ITERS   (OUT4_PER_B / (CHUNKS * BLOCK))   // 4 float4 per thread

// ---- CDNA5 async Global<->LDS DMA path (ASYNCcnt) --------------------------
#if defined(__gfx1250__) && \
    __has_builtin(__builtin_amdgcn_global_load_async_to_lds_b128) && \
    __has_builtin(__builtin_amdgcn_global_store_async_from_lds_b128)
#define USE_ASYNC 1
#else
#define USE_ASYNC 0
#endif

typedef int v4i __attribute__((ext_vector_type(4)));
typedef __attribute__((address_space(1))) v4i gv4i;  // global int4
typedef __attribute__((address_space(3))) v4i lv4i;  // LDS int4

// Flat global addresses are identical to AS1 addresses on AMDGPU.
__device__ __forceinline__ gv4i* as_global(const void* p) {
    return (gv4i*)(unsigned long long)p;
}
// Low 32 bits of a flat shared-aperture address are the LDS byte offset
// (ISA 10.2); AS3 pointers are 32-bit.
__device__ __forceinline__ lv4i* as_lds(const void* p) {
    return (lv4i*)(unsigned)(unsigned long long)p;
}

__device__ __forceinline__ void wait_async_zero() {
#if defined(__gfx1250__)
#if __has_builtin(__builtin_amdgcn_s_wait_asynccnt)
    __builtin_amdgcn_s_wait_asynccnt(0);
#else
    asm volatile("s_wait_asynccnt 0x0" ::: "memory");
#endif
#endif
}

__global__ __launch_bounds__(BLOCK)
void routing_mask_gather(const float4* __restrict__ in4,
                         const float*  __restrict__ routing,
                         float4* __restrict__ out4) {
    const int b = blockIdx.y;

    // argmax over 4 logits; strict '>' keeps the first max (jnp.argmax tie rule)
    const float r0 = routing[b * ROUTES + 0];
    const float r1 = routing[b * ROUTES + 1];
    const float r2 = routing[b * ROUTES + 2];
    const float r3 = routing[b * ROUTES + 3];
    int route = 0; float best = r0;
    if (r1 > best) { best = r1; route = 1; }
    if (r2 > best) { best = r2; route = 2; }
    if (r3 > best) { best = r3; route = 3; }

    const int tid = threadIdx.x;
    const long inBase  = (long)b * IN4_PER_B + (long)route * RW4;
    const long outBase = (long)b * OUT4_PER_B;

#if USE_ASYNC
    // DMA copy: global -> LDS -> global, no VGPR data movement.
    // Each thread owns a private 16B LDS slot, so a per-wave
    // s_wait_asynccnt(0) between the load and store phases is sufficient
    // (async ops track with ASYNCcnt; same-wave loads complete in order).
    __shared__ float4 smem[BLOCK * ITERS];   // 16 KB of the 320 KB WGP LDS

#pragma unroll
    for (int it = 0; it < ITERS; ++it) {
        const int o4 = (blockIdx.x * ITERS + it) * BLOCK + tid;  // [0, 32768)
        const int hw = o4 >> 5;          // o4 / RW4
        const int c4 = o4 & (RW4 - 1);   // o4 % RW4
        const float4* src = in4 + inBase + (long)hw * C4 + c4;
        __builtin_amdgcn_global_load_async_to_lds_b128(
            as_global(src), as_lds(&smem[it * BLOCK + tid]), 0, 0);
    }

    wait_async_zero();   // LDS now holds this wave's tiles

#pragma unroll
    for (int it = 0; it < ITERS; ++it) {
        const int o4 = (blockIdx.x * ITERS + it) * BLOCK + tid;
        float4* dst = out4 + outBase + o4;
        __builtin_amdgcn_global_store_async_from_lds_b128(
            as_global(dst), as_lds(&smem[it * BLOCK + tid]), 0, 0);
    }

    wait_async_zero();   // drain before wave retire
#else
    // Fallback: plain fully-coalesced b128 copy.
#pragma unroll
    for (int it = 0; it < ITERS; ++it) {
        const int o4 = (blockIdx.x * ITERS + it) * BLOCK + tid;
        const int hw = o4 >> 5;
        const int c4 = o4 & (RW4 - 1);
        out4[outBase + o4] = in4[inBase + (long)hw * C4 + c4];
    }
#endif
}

extern "C" void kernel_launch(void* const* d_in, const int* in_sizes, int n_in,
                              void* d_out, int out_size, void* d_ws, size_t ws_size,
                              hipStream_t stream) {
    (void)in_sizes; (void)n_in; (void)out_size; (void)d_ws; (void)ws_size;
    const float4* in4     = (const float4*)d_in[0];
    const float*  routing = (const float*)d_in[1];
    float4*       out4    = (float4*)d_out;

    dim3 grid(CHUNKS, Bn);   // 32 x 64 = 2048 blocks
    dim3 block(BLOCK);
    routing_mask_gather<<<grid, block, 0, stream>>>(in4, routing, out4);
}